// Gate_24343874633734
// MI455X (gfx1250) — compile-verified
//
#include <hip/hip_runtime.h>
#include <hip/hip_bf16.h>

typedef float v2f __attribute__((ext_vector_type(2)));
typedef float v4f __attribute__((ext_vector_type(4)));
typedef float v8f __attribute__((ext_vector_type(8)));

#define DIM        1024
#define NEXP       64
#define TOPK       8
#define TOK_BLK    128          // tokens per workgroup (8 waves x 16)
#define KC         32           // K chunk staged in LDS
#define XS_STRIDE  36           // 32 + pad: 144B rows -> 16B aligned + bank-conflict-free
#define WS_STRIDE  36
#define SC_STRIDE  68           // 64 + pad: 272B rows -> 16B aligned + bank-conflict-free

__global__ __launch_bounds__(256)
void gate_kernel(const float* __restrict__ x,
                 const float* __restrict__ w,
                 float* __restrict__ probs,
                 float* __restrict__ topv,
                 int*   __restrict__ topi)
{
    __shared__ float xs[2][TOK_BLK * XS_STRIDE];   // 2 x 18 KB
    __shared__ float wls[2][NEXP  * WS_STRIDE];    // 2 x  9 KB
    __shared__ float sc[TOK_BLK * SC_STRIDE];      //     34 KB

    const int tid  = threadIdx.x;
    const int wave = tid >> 5;
    const int lane = tid & 31;
    const int half = lane >> 4;       // 0: K=0..1, 1: K=2..3 (32-bit A/B layout)
    const int mr   = lane & 15;       // A row / B col within 16x16 tile
    const int w16  = wave * 16;       // wave's local token base
    const long tb  = (long)blockIdx.x * TOK_BLK;

    v8f acc0 = {}, acc1 = {}, acc2 = {}, acc3 = {};

    const int aoff  = (w16 + mr) * XS_STRIDE + half * 2;
    const int boff0 = ( 0 + mr) * WS_STRIDE + half * 2;
    const int boff1 = (16 + mr) * WS_STRIDE + half * 2;
    const int boff2 = (32 + mr) * WS_STRIDE + half * 2;
    const int boff3 = (48 + mr) * WS_STRIDE + half * 2;

    // per-thread staging assignments (row, col4) for x (4 vectors) and w (2 vectors)
    const int xr0 = (0 * 256 + tid) >> 3, xc0 = ((0 * 256 + tid) & 7) << 2;
    const int xr1 = (1 * 256 + tid) >> 3, xc1 = ((1 * 256 + tid) & 7) << 2;
    const int xr2 = (2 * 256 + tid) >> 3, xc2 = ((2 * 256 + tid) & 7) << 2;
    const int xr3 = (3 * 256 + tid) >> 3, xc3 = ((3 * 256 + tid) & 7) << 2;
    const int wr0 = (0 * 256 + tid) >> 3, wc0 = ((0 * 256 + tid) & 7) << 2;
    const int wr1 = (1 * 256 + tid) >> 3, wc1 = ((1 * 256 + tid) & 7) << 2;

    v4f rx0, rx1, rx2, rx3, rw0, rw1;

    // prologue: fetch chunk 0 and stage into buffer 0
    rx0 = __builtin_nontemporal_load((const v4f*)&x[(tb + xr0) * (long)DIM + xc0]);
    rx1 = __builtin_nontemporal_load((const v4f*)&x[(tb + xr1) * (long)DIM + xc1]);
    rx2 = __builtin_nontemporal_load((const v4f*)&x[(tb + xr2) * (long)DIM + xc2]);
    rx3 = __builtin_nontemporal_load((const v4f*)&x[(tb + xr3) * (long)DIM + xc3]);
    rw0 = *(const v4f*)&w[wr0 * (long)DIM + wc0];
    rw1 = *(const v4f*)&w[wr1 * (long)DIM + wc1];
    *(v4f*)&xs[0][xr0 * XS_STRIDE + xc0] = rx0;
    *(v4f*)&xs[0][xr1 * XS_STRIDE + xc1] = rx1;
    *(v4f*)&xs[0][xr2 * XS_STRIDE + xc2] = rx2;
    *(v4f*)&xs[0][xr3 * XS_STRIDE + xc3] = rx3;
    *(v4f*)&wls[0][wr0 * WS_STRIDE + wc0] = rw0;
    *(v4f*)&wls[0][wr1 * WS_STRIDE + wc1] = rw1;

    int cur = 0;
    for (int k0 = 0; k0 < DIM; k0 += KC) {
        __syncthreads();   // stage of buf `cur` visible; compute of buf `cur^1` finished
        const bool more = (k0 + KC) < DIM;
        const int  kn   = k0 + KC;
        if (more) {
            // issue next chunk's global loads now; latency hides under the WMMAs below
            rx0 = __builtin_nontemporal_load((const v4f*)&x[(tb + xr0) * (long)DIM + kn + xc0]);
            rx1 = __builtin_nontemporal_load((const v4f*)&x[(tb + xr1) * (long)DIM + kn + xc1]);
            rx2 = __builtin_nontemporal_load((const v4f*)&x[(tb + xr2) * (long)DIM + kn + xc2]);
            rx3 = __builtin_nontemporal_load((const v4f*)&x[(tb + xr3) * (long)DIM + kn + xc3]);
            rw0 = *(const v4f*)&w[wr0 * (long)DIM + kn + wc0];
            rw1 = *(const v4f*)&w[wr1 * (long)DIM + kn + wc1];
        }

        const float* __restrict__ xsc = xs[cur];
        const float* __restrict__ wlc = wls[cur];
        #pragma unroll
        for (int k = 0; k < KC; k += 4) {
            v2f a  = *(const v2f*)&xsc[aoff  + k];
            v2f b0 = *(const v2f*)&wlc[boff0 + k];
            v2f b1 = *(const v2f*)&wlc[boff1 + k];
            v2f b2 = *(const v2f*)&wlc[boff2 + k];
            v2f b3 = *(const v2f*)&wlc[boff3 + k];
            acc0 = __builtin_amdgcn_wmma_f32_16x16x4_f32(false, a, false, b0, (short)0, acc0, false, false);
            acc1 = __builtin_amdgcn_wmma_f32_16x16x4_f32(false, a, false, b1, (short)0, acc1, false, false);
            acc2 = __builtin_amdgcn_wmma_f32_16x16x4_f32(false, a, false, b2, (short)0, acc2, false, false);
            acc3 = __builtin_amdgcn_wmma_f32_16x16x4_f32(false, a, false, b3, (short)0, acc3, false, false);
        }

        if (more) {
            const int nxt = cur ^ 1;
            *(v4f*)&xs[nxt][xr0 * XS_STRIDE + xc0] = rx0;
            *(v4f*)&xs[nxt][xr1 * XS_STRIDE + xc1] = rx1;
            *(v4f*)&xs[nxt][xr2 * XS_STRIDE + xc2] = rx2;
            *(v4f*)&xs[nxt][xr3 * XS_STRIDE + xc3] = rx3;
            *(v4f*)&wls[nxt][wr0 * WS_STRIDE + wc0] = rw0;
            *(v4f*)&wls[nxt][wr1 * WS_STRIDE + wc1] = rw1;
        }
        cur ^= 1;
    }

    // scatter C/D layout into LDS: element r of acc -> token (r + 8*half), expert (16n + mr)
    #pragma unroll
    for (int r = 0; r < 8; ++r) {
        const int trow = (w16 + r + half * 8) * SC_STRIDE + mr;
        sc[trow +  0] = acc0[r];
        sc[trow + 16] = acc1[r];
        sc[trow + 32] = acc2[r];
        sc[trow + 48] = acc3[r];
    }

    // per-wave softmax + top-8: lanes 0..15 each own one of the wave's 16 tokens
    if (lane < 16) {
        const int  tl   = w16 + lane;
        const int  srow = tl * SC_STRIDE;
        const long tok  = tb + tl;

        float v[NEXP];
        #pragma unroll
        for (int e = 0; e < NEXP; e += 4) {
            v4f q = *(const v4f*)&sc[srow + e];
            v[e] = q.x; v[e + 1] = q.y; v[e + 2] = q.z; v[e + 3] = q.w;
        }
        float mx = v[0];
        #pragma unroll
        for (int e = 1; e < NEXP; ++e) mx = fmaxf(mx, v[e]);
        float sum = 0.0f;
        #pragma unroll
        for (int e = 0; e < NEXP; ++e) { float p = __expf(v[e] - mx); v[e] = p; sum += p; }
        const float inv = 1.0f / sum;
        #pragma unroll
        for (int e = 0; e < NEXP; ++e) v[e] *= inv;

        // normalized probs back to LDS for coalesced block-wide writeout
        #pragma unroll
        for (int e = 0; e < NEXP; e += 4) {
            v4f q = { v[e], v[e + 1], v[e + 2], v[e + 3] };
            *(v4f*)&sc[srow + e] = q;
        }

        // top-8, mutation-free (keeps v[] in VGPRs): order by (value desc, index asc)
        float lastv = 3.402823466e38f;
        int   laste = -1;
        #pragma unroll
        for (int j = 0; j < TOPK; ++j) {
            float bv = -3.402823466e38f;
            int   be = NEXP;
            #pragma unroll
            for (int e = 0; e < NEXP; ++e) {
                float val = v[e];
                bool elig   = (val < lastv) || ((val == lastv) && (e > laste));
                bool better = (val > bv)    || ((val == bv)    && (e < be));
                if (elig && better) { bv = val; be = e; }
            }
            __builtin_nontemporal_store(bv, &topv[tok * TOPK + j]);
            __builtin_nontemporal_store(be, &topi[tok * TOPK + j]);
            lastv = bv; laste = be;
        }
    }
    __syncthreads();

    // coalesced probs writeout: 128 tokens x 64 floats contiguous per block
    #pragma unroll
    for (int i = tid; i < TOK_BLK * (NEXP / 4); i += 256) {
        int t = i >> 4;
        int e = (i & 15) << 2;
        v4f q = *(const v4f*)&sc[t * SC_STRIDE + e];
        __builtin_nontemporal_store(q, (v4f*)&probs[(tb + t) * (long)NEXP + e]);
    }
}

extern "C" void kernel_launch(void* const* d_in, const int* in_sizes, int n_in,
                              void* d_out, int out_size, void* d_ws, size_t ws_size,
                              hipStream_t stream) {
    const float* x = (const float*)d_in[0];
    const float* w = (const float*)d_in[1];
    const long tokens = (long)in_sizes[0] / DIM;

    float* probs = (float*)d_out;
    float* topv  = probs + tokens * NEXP;
    int*   topi  = (int*)(topv + tokens * TOPK);

    const int blocks = (int)(tokens / TOK_BLK);
    gate_kernel<<<blocks, 256, 0, stream>>>(x, w, probs, topv, topi);
}